// RotationEmbed_27255862460554
// MI455X (gfx1250) — compile-verified
//
#include <hip/hip_runtime.h>
#include <stdint.h>

// ---------------------------------------------------------------------------
// out[i][0:16] = first columns of real Wigner-D blocks D^l(R_i), l=0..3.
// Closed form: col0 of D(R) = coefficients of Y_first(R^T p) in the SH basis.
// With a=R[:,0], b=R[:,1]:  everything is a fixed linear map W (16x32) applied
// to degree<=3 monomials of (a,b).  Per 16-item tile we do an actual matmul
// D(16x16) = F(16x32) x W^T(32x16) with v_wmma_f32_16x16x4_f32 (8 chunks).
// Data staged global->LDS with gfx1250 async-LDS loads, double-buffered.
// HBM-bound: 100 B/item * 500k = 50 MB -> ~2.1 us at 23.3 TB/s.
// ---------------------------------------------------------------------------

typedef float v2f __attribute__((ext_vector_type(2)));
typedef float v8f __attribute__((ext_vector_type(8)));
typedef int   i32x4 __attribute__((vector_size(4 * sizeof(int))));

#if defined(__gfx1250__) && __has_builtin(__builtin_amdgcn_global_load_async_to_lds_b128) && __has_builtin(__builtin_amdgcn_global_load_async_to_lds_b32)
#define HAVE_ASYNC_LDS 1
#else
#define HAVE_ASYNC_LDS 0
#endif

#if HAVE_ASYNC_LDS
// Param 1: typed AS(1) global pointer (int4* for b128, int* for b32),
// param 2: matching typed AS(3) LDS pointer (confirmed by compiler sigs).
typedef __attribute__((address_space(1))) i32x4 g_i32x4;
typedef __attribute__((address_space(3))) i32x4 l_i32x4;
typedef __attribute__((address_space(1))) int   g_i32;
typedef __attribute__((address_space(3))) int   l_i32;
#if __has_builtin(__builtin_amdgcn_s_wait_asynccnt)
#define WAIT_ASYNCCNT(n) __builtin_amdgcn_s_wait_asynccnt(n)
#else
#define WAIT_ASYNCCNT(n) asm volatile("s_wait_asynccnt %0" ::"i"(n) : "memory")
#endif
#else
#define WAIT_ASYNCCNT(n) do {} while (0)
#endif

// ---- constant projection matrix W^T: WT[k][n], k = monomial, n = output ----
#define SQ3  1.7320508075688772f   // sqrt(3)
#define CQ   1.2247448713915890f   // sqrt(3/2)
#define C2Q  2.4494897427831781f   // 2*CQ
#define CP   0.9682458365518543f   // sqrt(15)/4
#define C2P  1.9364916731037085f
#define C3P  2.9047375096555627f
#define CT   0.7905694150420949f   // sqrt(5/8)
#define C3T  2.3717082451262845f
#define CV   0.6123724356957945f   // CQ/2
#define C3V  1.8371173070873836f

__constant__ float WT[32][16] = {
/*k0 : 1      */ {1,0,0,0, 0,0,0,0,0, 0,0,0,0,0,0,0},
/*k1 : b1     */ {0,0,0,1, 0,0,0,0,0, 0,0,0,0,0,0,0},
/*k2 : b2     */ {0,1,0,0, 0,0,0,0,0, 0,0,0,0,0,0,0},
/*k3 : b3     */ {0,0,1,0, 0,0,0,0,0, 0,0,0,0,0,0,0},
/*k4 : a1b1   */ {0,0,0,0, 0,0,0,0,2, 0,0,0,0,0,0,0},
/*k5 : a1b2   */ {0,0,0,0, 1,0,0,0,0, 0,0,0,0,0,0,0},
/*k6 : a2b1   */ {0,0,0,0, 1,0,0,0,0, 0,0,0,0,0,0,0},
/*k7 : a2b3   */ {0,0,0,0, 0,1,0,0,0, 0,0,0,0,0,0,0},
/*k8 : a3b2   */ {0,0,0,0, 0,1,0,0,0, 0,0,0,0,0,0,0},
/*k9 : a3b3   */ {0,0,0,0, 0,0,SQ3,0,1, 0,0,0,0,0,0,0},
/*k10: a1b3   */ {0,0,0,0, 0,0,0,1,0, 0,0,0,0,0,0,0},
/*k11: a3b1   */ {0,0,0,0, 0,0,0,1,0, 0,0,0,0,0,0,0},
/*k12: a1a1b1 */ {0,0,0,0, 0,0,0,0,0, 0,0,0,0,-C3P,0,0.75f},
/*k13: b1b1b1 */ {0,0,0,0, 0,0,0,0,0, 0,0,0,0, CP,0,-0.25f},
/*k14: a2a2b2 */ {0,0,0,0, 0,0,0,0,0, -0.75f,0,-C3P,0,0,0,0},
/*k15: b2b2b2 */ {0,0,0,0, 0,0,0,0,0,  0.25f,0, CP,0,0,0,0},
/*k16: a3a3b3 */ {0,0,0,0, 0,0,0,0,0, 0,0,0, C3T,0, C3V,0},
/*k17: b3b3b3 */ {0,0,0,0, 0,0,0,0,0, 0,0,0,-CT,0,-CV,0},
/*k18: a1a1b2 */ {0,0,0,0, 0,0,0,0,0,  0.75f,0,-CP,0,0,0,0},
/*k19: b1b1b2 */ {0,0,0,0, 0,0,0,0,0, -0.75f,0, CP,0,0,0,0},
/*k20: a1a2b1 */ {0,0,0,0, 0,0,0,0,0,  1.5f,0,-C2P,0,0,0,0},
/*k21: a1a1b3 */ {0,0,0,0, 0,0,0,0,0, 0,0,0,0,0, CQ,0},
/*k22: b1b1b3 */ {0,0,0,0, 0,0,0,0,0, 0,0,0,0,0,-CQ,0},
/*k23: a1a3b1 */ {0,0,0,0, 0,0,0,0,0, 0,0,0,0,0,C2Q,0},
/*k24: a2a2b1 */ {0,0,0,0, 0,0,0,0,0, 0,0,0,0,-CP,0,-0.75f},
/*k25: b2b2b1 */ {0,0,0,0, 0,0,0,0,0, 0,0,0,0, CP,0, 0.75f},
/*k26: a1a2b2 */ {0,0,0,0, 0,0,0,0,0, 0,0,0,0,-C2P,0,-1.5f},
/*k27: a1a2b3 */ {0,0,0,0, 0,0,0,0,0, 0, CQ,0,0,0,0,0},
/*k28: b1b2b3 */ {0,0,0,0, 0,0,0,0,0, 0,-CQ,0,0,0,0,0},
/*k29: a1a3b2 */ {0,0,0,0, 0,0,0,0,0, 0, CQ,0,0,0,0,0},
/*k30: a2a3b1 */ {0,0,0,0, 0,0,0,0,0, 0, CQ,0,0,0,0,0},
/*k31: pad    */ {0,0,0,0, 0,0,0,0,0, 0,0,0,0,0,0,0},
};

// Stage one 16-item tile (144 dwords) into LDS.  Full tiles use the gfx1250
// async global->LDS path (1x b128 per lane + 1x b32 on the low half-wave);
// the (only possible) ragged tail tile falls back to a guarded copy.
__device__ __forceinline__ int stage_tile(const float* __restrict__ src,
                                          float* __restrict__ dst,
                                          int lane, int dwords_valid)
{
#if HAVE_ASYNC_LDS
    if (dwords_valid >= 144) {
        __builtin_amdgcn_global_load_async_to_lds_b128(
            (g_i32x4*)(void*)(src + 4 * lane),
            (l_i32x4*)(void*)(dst + 4 * lane), 0, 0);
        if (lane < 16)
            __builtin_amdgcn_global_load_async_to_lds_b32(
                (g_i32*)(void*)(src + 128 + lane),
                (l_i32*)(void*)(dst + 128 + lane), 0, 0);
        return 2;
    }
#endif
    for (int i = lane; i < 144; i += 32)
        dst[i] = (i < dwords_valid) ? src[i] : 0.0f;
    return 0;
}

__global__ __launch_bounds__(256) void
wigner_col0_wmma(const float* __restrict__ Rg, float* __restrict__ out, int nitems)
{
    __shared__ __align__(16) float lds[8][2][144];   // per-wave double buffer
    const int lane = threadIdx.x & 31;
    const int wid  = threadIdx.x >> 5;
    const int n    = lane & 15;   // output column (N) / item row (M)
    const int h    = lane >> 4;   // half-wave -> which K pair / which M half

    // B fragments (constant W^T), loaded once per wave and reused for every
    // tile.  B layout (K x N, f32): VGPR v, half h  <->  k = 2h + v, col = n.
    v2f vb[8];
#pragma unroll
    for (int c = 0; c < 8; ++c) {
        vb[c].x = WT[4 * c + 2 * h + 0][n];
        vb[c].y = WT[4 * c + 2 * h + 1][n];
    }

    const int wpb   = blockDim.x >> 5;
    const int gw    = gridDim.x * wpb;
    const int tiles = (nitems + 15) >> 4;
    int t = blockIdx.x * wpb + wid;
    if (t >= tiles) return;

    int buf = 0;
    stage_tile(Rg + (size_t)t * 144, &lds[wid][0][0], lane, nitems * 9 - t * 144);

    while (t < tiles) {
        const int tn = t + gw;
        int pend_next = 0;
        if (tn < tiles)    // prefetch next tile into the other buffer
            pend_next = stage_tile(Rg + (size_t)tn * 144, &lds[wid][buf ^ 1][0],
                                   lane, nitems * 9 - tn * 144);
#if HAVE_ASYNC_LDS
        if (pend_next) { WAIT_ASYNCCNT(2); } else { WAIT_ASYNCCNT(0); }
#endif
        asm volatile("" ::: "memory");

        // a = R[:,0], b = R[:,1]  (9-dword stride: conflict-free on 64 banks)
        const float* m9 = &lds[wid][buf][9 * n];
        const float a1 = m9[0], b1 = m9[1];
        const float a2 = m9[3], b2 = m9[4];
        const float a3 = m9[6], b3 = m9[7];

        float F[32];
        F[0]  = 1.0f;     F[1]  = b1;        F[2]  = b2;        F[3]  = b3;
        F[4]  = a1 * b1;  F[5]  = a1 * b2;   F[6]  = a2 * b1;   F[7]  = a2 * b3;
        F[8]  = a3 * b2;  F[9]  = a3 * b3;   F[10] = a1 * b3;   F[11] = a3 * b1;
        F[12] = a1 * F[4];   F[13] = b1 * b1 * b1;
        F[14] = a2 * a2 * b2; F[15] = b2 * b2 * b2;
        F[16] = a3 * F[9];   F[17] = b3 * b3 * b3;
        F[18] = a1 * F[5];   F[19] = b1 * b1 * b2;  F[20] = a2 * F[4];
        F[21] = a1 * F[10];  F[22] = b1 * b1 * b3;  F[23] = a3 * F[4];
        F[24] = a2 * F[6];   F[25] = b2 * b2 * b1;  F[26] = a2 * F[5];
        F[27] = a2 * F[10];  F[28] = b1 * b2 * b3;  F[29] = a3 * F[5];
        F[30] = a3 * F[6];   F[31] = 0.0f;

        // D(16x16) = F(16x32) x W^T(32x16)  as 8x  v_wmma_f32_16x16x4_f32.
        // A layout (M x K, f32): lane = M (mod 16), VGPR v, half h <-> k = 2h+v.
        v8f acc = {0.f, 0.f, 0.f, 0.f, 0.f, 0.f, 0.f, 0.f};
#pragma unroll
        for (int c = 0; c < 8; ++c) {
            v2f va;
            va.x = h ? F[4 * c + 2] : F[4 * c + 0];
            va.y = h ? F[4 * c + 3] : F[4 * c + 1];
            acc = __builtin_amdgcn_wmma_f32_16x16x4_f32(
                false, va, false, vb[c], (short)0, acc, false, false);
        }

        // C/D layout: VGPR j, half h  <->  row M = j + 8h, column n.
        // One base pointer + immediate offsets (j*64 B); uniform full-tile
        // fast path avoids 8 exec-mask branches in the steady state.
        const int limit = nitems - t * 16;
        float* o = out + ((size_t)t * 16 + 8 * h) * 16 + n;
        if (limit >= 16) {
#pragma unroll
            for (int j = 0; j < 8; ++j)
                __builtin_nontemporal_store(acc[j], o + j * 16);
        } else {
#pragma unroll
            for (int j = 0; j < 8; ++j)
                if (j + 8 * h < limit)
                    __builtin_nontemporal_store(acc[j], o + j * 16);
        }
        t = tn;
        buf ^= 1;
    }
}

extern "C" void kernel_launch(void* const* d_in, const int* in_sizes, int n_in,
                              void* d_out, int out_size, void* d_ws, size_t ws_size,
                              hipStream_t stream)
{
    (void)n_in; (void)out_size; (void)d_ws; (void)ws_size;
    const float* R   = (const float*)d_in[0];
    float*       out = (float*)d_out;
    const int nitems = in_sizes[0] / 9;          // 500000
    const int tiles  = (nitems + 15) >> 4;       // 31250 tiles of 16 items
    int blocks = (tiles + 7) / 8;                // 8 waves per 256-thread block
    if (blocks > 624) blocks = 624;              // ~5k waves, grid-stride loop
    if (blocks < 1)   blocks = 1;
    hipLaunchKernelGGL(wigner_col0_wmma, dim3(blocks), dim3(256), 0, stream,
                       R, out, nitems);
}